// Summarizer_56530359550542
// MI455X (gfx1250) — compile-verified
//
#include <hip/hip_runtime.h>
#include <math.h>

#define BB 32
#define SS 4096
#define HH 768
#define NN 128
#define TT 100
#define H4 (HH / 4)   // 192 float4 per row
#define LDSW 112      // T=100 padded to 7*16 (row stride in LDS)

typedef __attribute__((ext_vector_type(2))) float v2f;
typedef __attribute__((ext_vector_type(8))) float v8f;

// d_out layout (floats)
#define OFF_SCORES   0
#define OFF_MASK     4096
#define OFF_MEANDOC  8192
#define OFF_MEANSENT 32768
#define OFF_REC      3178496
#define OFF_TOPIC    3203072

// ------------------------------------------------------------------
// Kernel 1: one block per (b, segment). Coalesced float4 streaming of
// top_vec (the 402 MB bandwidth-bound pass). Writes segment sums to ws
// and mean_sent to d_out.
// ------------------------------------------------------------------
__global__ __launch_bounds__(192) void seg_reduce_kernel(
    const float* __restrict__ top_vec,
    const int*   __restrict__ clss,
    float*       __restrict__ segsum,     // ws (B,N,H)
    float*       __restrict__ mean_sent)  // d_out (B,N,H)
{
  const int blk = blockIdx.x;
  const int b = blk >> 7;          // / NN
  const int i = blk & (NN - 1);
  const int t = threadIdx.x;       // 0..191, one float4 column each

  const int end   = clss[b * NN + i];
  const int start = (i == 0) ? 0 : (clss[b * NN + i - 1] + 1);
  int cnt = end - start + 1;
  if (cnt < 1) cnt = 1;

  float ax = 0.f, ay = 0.f, az = 0.f, aw = 0.f;
  const float4* src =
      reinterpret_cast<const float4*>(top_vec) + ((size_t)b * SS + (size_t)start) * H4 + t;
  for (int s = start; s <= end; ++s) {
    float4 v = *src;
    ax += v.x; ay += v.y; az += v.z; aw += v.w;
    src += H4;
    __builtin_prefetch((const void*)src, 0, 1);   // global_prefetch_b8
  }
  const size_t o = ((size_t)(b * NN + i)) * H4 + t;
  reinterpret_cast<float4*>(segsum)[o] = make_float4(ax, ay, az, aw);
  const float inv = 1.0f / (float)cnt;
  reinterpret_cast<float4*>(mean_sent)[o] = make_float4(ax * inv, ay * inv, az * inv, aw * inv);
}

// ------------------------------------------------------------------
// Kernel 2: mean_doc[b,h] = (sum over segments of segsum) / (last+1).
// (doc mask == union of all segments, so this equals the masked mean)
// ------------------------------------------------------------------
__global__ __launch_bounds__(192) void mean_doc_kernel(
    const float* __restrict__ segsum,
    const int*   __restrict__ clss,
    float*       __restrict__ mean_doc)   // d_out (B,H)
{
  const int b = blockIdx.x;
  const int t = threadIdx.x;
  float ax = 0.f, ay = 0.f, az = 0.f, aw = 0.f;
  const float4* src = reinterpret_cast<const float4*>(segsum) + (size_t)b * NN * H4 + t;
  for (int i = 0; i < NN; ++i) {
    float4 v = src[(size_t)i * H4];
    ax += v.x; ay += v.y; az += v.z; aw += v.w;
  }
  const int last = clss[b * NN + NN - 1];
  const float inv = 1.0f / (float)(last + 1);
  reinterpret_cast<float4*>(mean_doc)[(size_t)b * H4 + t] =
      make_float4(ax * inv, ay * inv, az * inv, aw * inv);
}

// ------------------------------------------------------------------
// Kernel 3: topic pipeline, single workgroup, V_WMMA_F32_16X16X4_F32.
//   logits(32x100) = mean_doc(32x768) @ W_hid(768x100) + b_hid
//   dist = softmax(logits)          (rows in LDS)
//   rec_emb(32x768) = dist @ topic_emb(100x768)
// A 16x4 f32 layout: lane l -> M=l%16, VGPRv -> K = k + v + 2*(l/16).
// B/D: rows striped across lanes within a VGPR.
// Guards are branch-free (clamped address + 0/1 multiply) so the inner
// loops are straight-line load -> wmma with EXEC untouched.
// ------------------------------------------------------------------
__global__ __launch_bounds__(256) void topic_kernel(
    const float* __restrict__ mean_doc,   // (32,768)
    const float* __restrict__ W_hid,      // (768,100)
    const float* __restrict__ b_hid,      // (100)
    const float* __restrict__ topic_emb,  // (100,768)
    float*       __restrict__ rec_emb)    // (32,768)
{
  __shared__ float Lm[BB * LDSW];
  const int tid   = threadIdx.x;
  const int wave  = tid >> 5;
  const int lane  = tid & 31;
  const int lhalf = lane >> 4;   // 0 or 1
  const int lmod  = lane & 15;

  // ---- Phase A: logits -> LDS. Waves 0..6 each own one 16-col N tile.
  if (wave < 7) {
    const int n = wave * 16 + lmod;
    const bool  nvalid = (n < TT);
    const float nmaskf = nvalid ? 1.0f : 0.0f;   // loop-invariant lane mask
    const int   nsafe  = nvalid ? n : 0;         // clamped (always in-bounds)
    for (int mt = 0; mt < 2; ++mt) {
      const float* arow = mean_doc + (mt * 16 + lmod) * HH;
      v8f acc = {};
      for (int k = 0; k < HH; k += 4) {
        const int ka = k + 2 * lhalf;
        v2f a, bm;
        a.x  = arow[ka];
        a.y  = arow[ka + 1];
        bm.x = W_hid[(ka    ) * TT + nsafe] * nmaskf;
        bm.y = W_hid[(ka + 1) * TT + nsafe] * nmaskf;
        acc = __builtin_amdgcn_wmma_f32_16x16x4_f32(
            false, a, false, bm, (short)0, acc, false, false);
      }
      const float bias = b_hid[nsafe];
#pragma unroll
      for (int v = 0; v < 8; ++v) {
        const int m = mt * 16 + v + 8 * lhalf;
        Lm[m * LDSW + n] = nvalid ? (acc[v] + bias) : -1e30f;
      }
    }
  }
  __syncthreads();

  // ---- Phase B: row softmax over 100 topics; zero the pad columns.
  if (tid < BB) {
    float mx = -1e30f;
    for (int n = 0; n < TT; ++n) mx = fmaxf(mx, Lm[tid * LDSW + n]);
    float s = 0.0f;
    for (int n = 0; n < TT; ++n) {
      float e = expf(Lm[tid * LDSW + n] - mx);
      Lm[tid * LDSW + n] = e;
      s += e;
    }
    const float inv = 1.0f / s;
    for (int n = 0; n < TT; ++n) Lm[tid * LDSW + n] *= inv;
    for (int n = TT; n < LDSW; ++n) Lm[tid * LDSW + n] = 0.0f;
  }
  __syncthreads();

  // ---- Phase C: rec_emb = dist @ topic_emb.  2 M-tiles x 48 N-tiles.
  // k = 0..96 step 4 covers K = 0..99 exactly (each x4 WMMA consumes
  // k..k+3), so no guards and no OOB reads of topic_emb are needed.
  for (int tile = wave; tile < 96; tile += 8) {
    const int mt = tile & 1;
    const int nt = tile >> 1;
    const int nc = nt * 16 + lmod;              // output column, < 768
    const float* arow = &Lm[(mt * 16 + lmod) * LDSW];
    v8f acc = {};
    for (int k = 0; k <= TT - 4; k += 4) {      // 25 iterations, K=0..99
      const int ka = k + 2 * lhalf;
      v2f a, bm;
      a.x  = arow[ka];
      a.y  = arow[ka + 1];
      bm.x = topic_emb[(ka    ) * HH + nc];
      bm.y = topic_emb[(ka + 1) * HH + nc];
      acc = __builtin_amdgcn_wmma_f32_16x16x4_f32(
          false, a, false, bm, (short)0, acc, false, false);
    }
#pragma unroll
    for (int v = 0; v < 8; ++v) {
      const int m = mt * 16 + v + 8 * lhalf;
      rec_emb[m * HH + nc] = acc[v];
    }
  }
}

// ------------------------------------------------------------------
// Kernel 4: one wave32 per (b,i):
//   score = sigmoid(mean_sent[b,i]·w_cls + rec_emb[b]·w_top + b_cls)*mask
// ------------------------------------------------------------------
__global__ __launch_bounds__(256) void score_kernel(
    const float* __restrict__ mean_sent,
    const float* __restrict__ rec_emb,
    const float* __restrict__ w_cls,
    const float* __restrict__ w_top,
    const float* __restrict__ b_cls,
    const unsigned char* __restrict__ mask,
    float* __restrict__ sent_scores,
    float* __restrict__ mask_out)
{
  const int wave = threadIdx.x >> 5;
  const int lane = threadIdx.x & 31;
  const int g = blockIdx.x * 8 + wave;     // 0..4095 = (b,i)
  const int b = g >> 7;

  const float* ms = mean_sent + (size_t)g * HH;
  const float* re = rec_emb   + (size_t)b * HH;
  float p = 0.0f;
  for (int h = lane; h < HH; h += 32)
    p += ms[h] * w_cls[h] + re[h] * w_top[h];
#pragma unroll
  for (int m = 16; m >= 1; m >>= 1)
    p += __shfl_xor(p, m, 32);

  if (lane == 0) {
    const float mk = mask[g] ? 1.0f : 0.0f;
    const float z = 1.0f / (1.0f + expf(-(p + b_cls[0])));
    sent_scores[g] = z * mk;
    mask_out[g] = mk;
  }
}

extern "C" void kernel_launch(void* const* d_in, const int* in_sizes, int n_in,
                              void* d_out, int out_size, void* d_ws, size_t ws_size,
                              hipStream_t stream) {
  const float* top_vec    = (const float*)d_in[0];
  const int*   clss       = (const int*)d_in[1];
  const unsigned char* mask_cls = (const unsigned char*)d_in[2];
  const float* W_hid      = (const float*)d_in[3];
  const float* b_hid      = (const float*)d_in[4];
  const float* topic_emb  = (const float*)d_in[5];
  const float* w_cls      = (const float*)d_in[6];
  const float* w_top      = (const float*)d_in[7];
  const float* b_cls      = (const float*)d_in[8];

  float* out    = (float*)d_out;
  float* segsum = (float*)d_ws;   // needs B*N*H*4 = 12.6 MB

  seg_reduce_kernel<<<BB * NN, 192, 0, stream>>>(
      top_vec, clss, segsum, out + OFF_MEANSENT);

  mean_doc_kernel<<<BB, 192, 0, stream>>>(
      segsum, clss, out + OFF_MEANDOC);

  topic_kernel<<<1, 256, 0, stream>>>(
      out + OFF_MEANDOC, W_hid, b_hid, topic_emb, out + OFF_REC);

  score_kernel<<<(BB * NN) / 8, 256, 0, stream>>>(
      out + OFF_MEANSENT, out + OFF_REC, w_cls, w_top, b_cls, mask_cls,
      out + OFF_SCORES, out + OFF_MASK);

  hipMemcpyAsync(out + OFF_TOPIC, topic_emb, (size_t)TT * HH * sizeof(float),
                 hipMemcpyDeviceToDevice, stream);
}